// MixHop_47107201303138
// MI455X (gfx1250) — compile-verified
//
#include <hip/hip_runtime.h>
#include <hip/hip_bf16.h>

#define NN 50000
#define EE 1600000
#define BN_EPSF 1e-5f

typedef __attribute__((ext_vector_type(2))) float v2f;
typedef __attribute__((ext_vector_type(8))) float v8f;

// ---------------- gcn_norm: degree -> d^-1/2 ----------------
__global__ void k_deg_init(float* __restrict__ deg) {
  int i = blockIdx.x * blockDim.x + threadIdx.x;
  if (i < NN) deg[i] = 1.0f;  // self-loop contributes 1 to every node's degree
}
__global__ void k_deg_accum(const int* __restrict__ col, float* __restrict__ deg) {
  int e = blockIdx.x * blockDim.x + threadIdx.x;
  if (e < EE) atomicAdd(&deg[col[e]], 1.0f);
}
__global__ void k_deg_to_dinv(float* __restrict__ deg) {
  int i = blockIdx.x * blockDim.x + threadIdx.x;
  if (i < NN) { float d = deg[i]; deg[i] = d > 0.f ? rsqrtf(d) : 0.f; }
}

// ---------------- SpMM: out[row] += dinv[row]*dinv[col] * in[col] ----------------
// diag kernel writes the self-loop term and fully initializes the output block
__global__ void k_spmm_diag(const float* __restrict__ in, const float* __restrict__ dinv,
                            float* __restrict__ out, int ldo, int coff) {
  long long t = (long long)blockIdx.x * blockDim.x + threadIdx.x;
  if (t >= (long long)NN * 128) return;
  int n = (int)(t >> 7), c = (int)(t & 127);
  float d = dinv[n];
  out[(long long)n * ldo + coff + c] = d * d * in[(long long)n * 128 + c];
}
__global__ void k_spmm_edges(const int* __restrict__ row, const int* __restrict__ col,
                             const float* __restrict__ dinv, const float* __restrict__ in,
                             float* __restrict__ out, int ldo, int coff) {
  long long t = (long long)blockIdx.x * blockDim.x + threadIdx.x;
  if (t >= (long long)EE * 128) return;
  int e = (int)(t >> 7), c = (int)(t & 127);
  int r = row[e], s = col[e];
  float v = dinv[r] * dinv[s];
  atomicAdd(&out[(long long)r * ldo + coff + c], v * in[(long long)s * 128 + c]);
}

// ---------------- dense GEMM: C[,coff:] (=|+=) A(MxK) @ B(Kx128) + bias ----------------
// 8 waves per block, one 16x16 tile per wave, V_WMMA_F32_16X16X4_F32 K-steps.
__global__ __launch_bounds__(256) void k_wmma_gemm(
    const float* __restrict__ A, int lda,
    const float* __restrict__ B, int ldb,
    const float* __restrict__ bias,
    float* __restrict__ C, int ldc, int coff,
    int K, int accum)
{
  const int lane = threadIdx.x & 31;
  const int wave = threadIdx.x >> 5;
  const int m0 = blockIdx.x * 16;   // 50000/16 = 3125 blocks exactly -> EXEC all-ones
  const int n0 = wave * 16;         // 8 waves cover N=128
  const int nl = lane & 15;
  const int hi = lane >> 4;         // lane half selects K pair (A) / M half (C)

  float* cptr = C + (long long)(m0 + hi * 8) * ldc + coff + n0 + nl;
  v8f acc;
  if (accum) {
#pragma unroll
    for (int i = 0; i < 8; ++i) acc[i] = cptr[(long long)i * ldc];
  } else {
#pragma unroll
    for (int i = 0; i < 8; ++i) acc[i] = 0.f;
  }

  const float* arow  = A + (long long)(m0 + nl) * lda + hi * 2;        // K pair contiguous
  const float* bbase = B + (long long)(hi * 2) * ldb + n0 + nl;

#if __has_builtin(__builtin_amdgcn_wmma_f32_16x16x4_f32)
  for (int kk = 0; kk < K; kk += 4) {
    v2f a = *(const v2f*)(arow + kk);                 // global_load_b64
    const float* bp = bbase + (long long)kk * ldb;
    v2f b;
    b.x = bp[0];
    b.y = bp[ldb];
    acc = __builtin_amdgcn_wmma_f32_16x16x4_f32(false, a, false, b,
                                                (short)0, acc, false, false);
  }
#else
  // scalar fallback (keeps file compiling if the f32 builtin signature differs)
  for (int i = 0; i < 8; ++i) {
    int m = m0 + hi * 8 + i, n = n0 + nl;
    float s = acc[i];
    for (int k = 0; k < K; ++k)
      s += A[(long long)m * lda + k] * B[(long long)k * ldb + n];
    acc[i] = s;
  }
#endif

  float bv = bias ? bias[n0 + nl] : 0.f;
#pragma unroll
  for (int i = 0; i < 8; ++i) cptr[(long long)i * ldc] = acc[i] + bv;
}

// ---------------- BatchNorm (batch stats, biased var) + ReLU ----------------
__global__ __launch_bounds__(256) void k_bn_stats(const float* __restrict__ H,
                                                  float* __restrict__ stats) {
  __shared__ float ssum[256];
  __shared__ float ssq[256];
  int c = blockIdx.x;  // 0..383
  float s = 0.f, q = 0.f;
  for (int r = threadIdx.x; r < NN; r += 256) {
    float v = H[(long long)r * 384 + c];
    s += v; q += v * v;
  }
  ssum[threadIdx.x] = s; ssq[threadIdx.x] = q;
  __syncthreads();
  for (int w = 128; w > 0; w >>= 1) {
    if (threadIdx.x < w) {
      ssum[threadIdx.x] += ssum[threadIdx.x + w];
      ssq[threadIdx.x]  += ssq[threadIdx.x + w];
    }
    __syncthreads();
  }
  if (threadIdx.x == 0) {
    float mean = ssum[0] / (float)NN;
    stats[c] = mean;
    stats[384 + c] = ssq[0] / (float)NN - mean * mean;
  }
}
__global__ void k_bn_relu(float* __restrict__ H, const float* __restrict__ stats,
                          const float* __restrict__ gamma, const float* __restrict__ beta) {
  long long t = (long long)blockIdx.x * blockDim.x + threadIdx.x;
  if (t >= (long long)NN * 384) return;
  int c = (int)(t % 384);
  float v = (H[t] - stats[c]) * rsqrtf(stats[384 + c] + BN_EPSF) * gamma[c] + beta[c];
  H[t] = v > 0.f ? v : 0.f;
}
__global__ void k_out_init(float* __restrict__ out, const float* __restrict__ b) {
  long long t = (long long)blockIdx.x * blockDim.x + threadIdx.x;
  if (t >= (long long)NN * 128) return;
  out[t] = b[t & 127];
}

extern "C" void kernel_launch(void* const* d_in, const int* in_sizes, int n_in,
                              void* d_out, int out_size, void* d_ws, size_t ws_size,
                              hipStream_t stream) {
  (void)in_sizes; (void)n_in; (void)out_size; (void)ws_size;
  const float* x   = (const float*)d_in[0];
  const int*   ei  = (const int*)d_in[1];
  const float* l1w = (const float*)d_in[2];
  const float* l1b = (const float*)d_in[3];
  const float* gam = (const float*)d_in[4];
  const float* bet = (const float*)d_in[5];
  const float* l2w = (const float*)d_in[6];
  const float* l2b = (const float*)d_in[7];
  const float* fpw = (const float*)d_in[8];
  const float* fpb = (const float*)d_in[9];
  float* out = (float*)d_out;

  const int* row = ei;       // edge_index[0]
  const int* col = ei + EE;  // edge_index[1]

  // workspace layout (floats): dinv | t0 | t1 | H1 | stats  (~128.3 MB)
  float* ws = (float*)d_ws;
  long long o = 0;
  float* dinv = ws;               o += ((NN + 63) / 64) * 64;
  float* t0   = ws + o;           o += (long long)NN * 128;
  float* t1   = ws + o;           o += (long long)NN * 128;
  float* H1   = ws + o;           o += (long long)NN * 384;
  float* stats = ws + o;          o += 768;

  const int B = 256;
  const int gN    = (NN + B - 1) / B;
  const int gE    = (EE + B - 1) / B;
  const int gN128 = (int)(((long long)NN * 128 + B - 1) / B);
  const int gN384 = (int)(((long long)NN * 384 + B - 1) / B);
  const int gE128 = (int)(((long long)EE * 128 + B - 1) / B);
  const int MT = NN / 16;  // 3125

  // gcn_norm
  k_deg_init   <<<gN, B, 0, stream>>>(dinv);
  k_deg_accum  <<<gE, B, 0, stream>>>(col, dinv);
  k_deg_to_dinv<<<gN, B, 0, stream>>>(dinv);

  // ---- layer 1: H1 = [x@W0+b0 | A(x@W1+b1) | A^2(x@W2+b2)] ----
  k_wmma_gemm<<<MT, B, 0, stream>>>(x, 128, l1w,               128, l1b,       H1, 384, 0,   128, 0);
  k_wmma_gemm<<<MT, B, 0, stream>>>(x, 128, l1w + 128 * 128,   128, l1b + 128, t0, 128, 0,   128, 0);
  k_spmm_diag <<<gN128, B, 0, stream>>>(t0, dinv, H1, 384, 128);
  k_spmm_edges<<<gE128, B, 0, stream>>>(row, col, dinv, t0, H1, 384, 128);
  k_wmma_gemm<<<MT, B, 0, stream>>>(x, 128, l1w + 2 * 128 * 128, 128, l1b + 256, t0, 128, 0, 128, 0);
  k_spmm_diag <<<gN128, B, 0, stream>>>(t0, dinv, t1, 128, 0);
  k_spmm_edges<<<gE128, B, 0, stream>>>(row, col, dinv, t0, t1, 128, 0);
  k_spmm_diag <<<gN128, B, 0, stream>>>(t1, dinv, H1, 384, 256);
  k_spmm_edges<<<gE128, B, 0, stream>>>(row, col, dinv, t1, H1, 384, 256);

  // ---- BatchNorm + ReLU (in place on H1) ----
  k_bn_stats<<<384, B, 0, stream>>>(H1, stats);
  k_bn_relu <<<gN384, B, 0, stream>>>(H1, stats, gam, bet);

  // ---- out = fp_b, then stream layer-2 hops through accumulating GEMM ----
  k_out_init<<<gN128, B, 0, stream>>>(out, fpb);

  // j = 0: out += (H1@W0+b0) @ fp_w[0:128]
  k_wmma_gemm<<<MT, B, 0, stream>>>(H1, 384, l2w,             128, l2b,       t0, 128, 0, 384, 0);
  k_wmma_gemm<<<MT, B, 0, stream>>>(t0, 128, fpw,             128, nullptr,  out, 128, 0, 128, 1);
  // j = 1: out += A(H1@W1+b1) @ fp_w[128:256]
  k_wmma_gemm<<<MT, B, 0, stream>>>(H1, 384, l2w + 384 * 128, 128, l2b + 128, t0, 128, 0, 384, 0);
  k_spmm_diag <<<gN128, B, 0, stream>>>(t0, dinv, t1, 128, 0);
  k_spmm_edges<<<gE128, B, 0, stream>>>(row, col, dinv, t0, t1, 128, 0);
  k_wmma_gemm<<<MT, B, 0, stream>>>(t1, 128, fpw + 128 * 128, 128, nullptr,  out, 128, 0, 128, 1);
  // j = 2: out += A^2(H1@W2+b2) @ fp_w[256:384]
  k_wmma_gemm<<<MT, B, 0, stream>>>(H1, 384, l2w + 2 * 384 * 128, 128, l2b + 256, t0, 128, 0, 384, 0);
  k_spmm_diag <<<gN128, B, 0, stream>>>(t0, dinv, t1, 128, 0);
  k_spmm_edges<<<gE128, B, 0, stream>>>(row, col, dinv, t0, t1, 128, 0);
  k_spmm_diag <<<gN128, B, 0, stream>>>(t1, dinv, t0, 128, 0);
  k_spmm_edges<<<gE128, B, 0, stream>>>(row, col, dinv, t1, t0, 128, 0);
  k_wmma_gemm<<<MT, B, 0, stream>>>(t0, 128, fpw + 2 * 128 * 128, 128, nullptr, out, 128, 0, 128, 1);
}